// GraphNetwork2_35003983462588
// MI455X (gfx1250) — compile-verified
//
#include <hip/hip_runtime.h>
#include <hip/hip_bf16.h>
#include <math.h>

#define N_NODES_C 100000
#define N_EDGES_C 1600000
#define N_GRAPHS_C 64
#define CH 64
#define NCLS 10
#define BN_EPS 1e-5f

typedef float v2f __attribute__((ext_vector_type(2)));
typedef float v8f __attribute__((ext_vector_type(8)));

// ---------------- weight prep: Wcat[64 x 128] = [Wtop - Wbot | Wbot] ----------------
__global__ void prep_w_kernel(const float* __restrict__ W, float* __restrict__ Wcat) {
  int idx = blockIdx.x * blockDim.x + threadIdx.x;
  if (idx >= 64 * 128) return;
  int k = idx >> 7, n = idx & 127;
  float v;
  if (n < 64) v = W[k * 64 + n] - W[(k + 64) * 64 + n];
  else        v = W[(k + 64) * 64 + (n - 64)];
  Wcat[idx] = v;
}

// ---------------- node GEMM: AB[N,128] = X[N,64] @ Wcat[64,128] via WMMA f32 16x16x4 ----------------
// A frag (16x4 f32): lane l<16 holds X[m=l, k=4kk+{0,1}], lane l+16 holds X[m=l, k=4kk+{2,3}]
// B frag (4x16 f32): vgpr v, lanehalf h: row K = 4kk + 2h + v, col = lane&15
// C/D (16x16 f32):  vgpr i, lanes0-15 -> M=i, lanes16-31 -> M=i+8, N = lane&15
__global__ void node_gemm_kernel(const float* __restrict__ X, const float* __restrict__ Wc,
                                 float* __restrict__ AB) {
  int wave = threadIdx.x >> 5;   // 0..7 : which 16-wide column tile of the 128 outputs
  int lane = threadIdx.x & 31;
  int m = lane & 15;
  int h = lane >> 4;
  int n = (wave << 4) + m;       // global output column for B/C
  int row0 = blockIdx.x << 4;    // 16 node rows per block-tile (100000/16 = 6250 exact)

  v2f bf[16];
#pragma unroll
  for (int kk = 0; kk < 16; ++kk) {
    int k0 = (kk << 2) + (h << 1);
    bf[kk].x = Wc[(k0    ) * 128 + n];
    bf[kk].y = Wc[(k0 + 1) * 128 + n];
  }

  v8f acc = {};
  const float* xr = X + (size_t)(row0 + m) * 64 + (h << 1);
#pragma unroll
  for (int kk = 0; kk < 16; ++kk) {
    v2f a;
    a.x = xr[(kk << 2)];
    a.y = xr[(kk << 2) + 1];
    acc = __builtin_amdgcn_wmma_f32_16x16x4_f32(false, a, false, bf[kk], (short)0, acc,
                                                false, false);
  }

  float* out = AB + (size_t)row0 * 128 + (wave << 4) + m;
#pragma unroll
  for (int i = 0; i < 8; ++i) {
    int r = i + (h << 3);
    out[(size_t)r * 128] = acc[i];
  }
}

// ---------------- zero helper ----------------
__global__ void zero_kernel(unsigned int* __restrict__ p, int n) {
  for (int i = blockIdx.x * blockDim.x + threadIdx.x; i < n; i += gridDim.x * blockDim.x)
    p[i] = 0u;
}

// ---------------- order-preserving float <-> uint bijection ----------------
// fenc is strictly increasing; key 0 decodes to NaN, unreachable for real data -> "no edge" sentinel
__device__ __forceinline__ unsigned int fenc(float f) {
  int i = __float_as_int(f);
  return (unsigned int)(i ^ ((i >> 31) | 0x80000000));
}
__device__ __forceinline__ float fdec(unsigned int u) {
  int i = (int)u;
  return __int_as_float(i ^ ((~(i >> 31)) | 0x80000000));
}

// ---------------- segment-max of B[src] into key buffer M[dst] ----------------
// 16 lanes per edge, float4 per lane -> one uniform global_atomic_max_u32 per channel
__global__ void edge_max_kernel(const int* __restrict__ src, const int* __restrict__ dst,
                                const float* __restrict__ AB, unsigned int* __restrict__ M,
                                int E) {
  int e = blockIdx.x * (blockDim.x >> 4) + (threadIdx.x >> 4);  // 16 edges per 256-thr block
  if (e >= E) return;
  int c4 = (threadIdx.x & 15) << 2;
  int s = src[e];
  int d = dst[e];
  const float4 v = *(const float4*)(AB + (size_t)s * 128 + 64 + c4);
  unsigned int* mp = M + (size_t)d * 64 + c4;
  atomicMax(mp + 0, fenc(v.x));
  atomicMax(mp + 1, fenc(v.y));
  atomicMax(mp + 2, fenc(v.z));
  atomicMax(mp + 3, fenc(v.w));
}

// ---------------- combine: h_pre = relu(A + b + max) (0 for edge-less nodes), in-place into AB,
//                  fused BN sum / sumsq accumulation ----------------
__global__ void combine_stats_kernel(float* __restrict__ AB, const unsigned int* __restrict__ M,
                                     const float* __restrict__ bias,
                                     float* __restrict__ ssum, float* __restrict__ ssq, int N) {
  __shared__ float ls[64], lq[64];
  int t = threadIdx.x;
  if (t < 64) { ls[t] = 0.0f; lq[t] = 0.0f; }
  __syncthreads();
  int c = t & 63;
  float b = bias[c];
  float lsum = 0.0f, lsq = 0.0f;
  int rpb = blockDim.x >> 6;  // 4 nodes per iteration per block
  for (int n = blockIdx.x * rpb + (t >> 6); n < N; n += gridDim.x * rpb) {
    float a = AB[(size_t)n * 128 + c];
    unsigned int key = M[(size_t)n * 64 + c];
    float hp = (key != 0u) ? fmaxf(a + b + fdec(key), 0.0f) : 0.0f;
    AB[(size_t)n * 128 + c] = hp;
    lsum += hp;
    lsq += hp * hp;
  }
  atomicAdd(&ls[c], lsum);
  atomicAdd(&lq[c], lsq);
  __syncthreads();
  if (t < 64) { atomicAdd(&ssum[t], ls[t]); atomicAdd(&ssq[t], lq[t]); }
}

// ---------------- BN stat finalize: scale/shift per channel ----------------
__global__ void bn_finalize_kernel(const float* __restrict__ ssum, const float* __restrict__ ssq,
                                   const float* __restrict__ g, const float* __restrict__ be,
                                   float* __restrict__ scale, float* __restrict__ shift, int N) {
  int c = threadIdx.x;
  if (c >= 64) return;
  float inv = 1.0f / (float)N;
  float mu = ssum[c] * inv;
  float var = fmaxf(ssq[c] * inv - mu * mu, 0.0f);
  float sc = g[c] * rsqrtf(var + BN_EPS);
  scale[c] = sc;
  shift[c] = be[c] - mu * sc;
}

// ---------------- BN apply + relu: H = relu(h_pre*scale + shift) ----------------
__global__ void bn_relu_kernel(const float* __restrict__ AB, const float* __restrict__ scale,
                               const float* __restrict__ shift, float* __restrict__ H, int N) {
  int idx = blockIdx.x * blockDim.x + threadIdx.x;
  if (idx >= N * 64) return;
  int n = idx >> 6, c = idx & 63;
  float v = AB[(size_t)n * 128 + c] * scale[c] + shift[c];
  H[idx] = fmaxf(v, 0.0f);
}

// ---------------- per-graph sum pooling of h_pre (BN affine folded in later) ----------------
__global__ void pool_kernel(const float* __restrict__ AB, const int* __restrict__ batch,
                            float* __restrict__ gsum, float* __restrict__ gcnt, int N) {
  int t = threadIdx.x;
  int c = t & 63;
  int rpb = blockDim.x >> 6;
  for (int n = blockIdx.x * rpb + (t >> 6); n < N; n += gridDim.x * rpb) {
    int g = batch[n];
    atomicAdd(&gsum[(size_t)g * 64 + c], AB[(size_t)n * 128 + c]);
    if (c == 0) atomicAdd(&gcnt[g], 1.0f);
  }
}

// ---------------- head: pooled = (scale*sum + shift*cnt)/max(cnt,1); logits; softmax ----------------
__global__ void head_kernel(const float* __restrict__ gsum, const float* __restrict__ gcnt,
                            const float* __restrict__ scale, const float* __restrict__ shift,
                            const float* __restrict__ Wo, const float* __restrict__ bo,
                            float* __restrict__ out) {
  int g = threadIdx.x;
  if (g >= N_GRAPHS_C) return;
  float cnt = gcnt[g];
  float inv = 1.0f / fmaxf(cnt, 1.0f);
  float logit[NCLS];
#pragma unroll
  for (int k = 0; k < NCLS; ++k) logit[k] = bo[k];
  for (int c = 0; c < 64; ++c) {
    float pooled = (scale[c] * gsum[(size_t)g * 64 + c] + shift[c] * cnt) * inv;
#pragma unroll
    for (int k = 0; k < NCLS; ++k) logit[k] += pooled * Wo[c * NCLS + k];
  }
  float mx = logit[0];
#pragma unroll
  for (int k = 1; k < NCLS; ++k) mx = fmaxf(mx, logit[k]);
  float se = 0.0f;
#pragma unroll
  for (int k = 0; k < NCLS; ++k) { logit[k] = __expf(logit[k] - mx); se += logit[k]; }
  float is = 1.0f / se;
#pragma unroll
  for (int k = 0; k < NCLS; ++k) out[g * NCLS + k] = logit[k] * is;
}

extern "C" void kernel_launch(void* const* d_in, const int* in_sizes, int n_in,
                              void* d_out, int out_size, void* d_ws, size_t ws_size,
                              hipStream_t stream) {
  const float* x  = (const float*)d_in[0];
  const int* ei   = (const int*)d_in[1];
  const int* batch = (const int*)d_in[2];
  const float* W1 = (const float*)d_in[3];
  const float* b1 = (const float*)d_in[4];
  const float* W2 = (const float*)d_in[5];
  const float* b2 = (const float*)d_in[6];
  const float* W3 = (const float*)d_in[7];
  const float* b3 = (const float*)d_in[8];
  const float* g1 = (const float*)d_in[9];
  const float* be1 = (const float*)d_in[10];
  const float* g2 = (const float*)d_in[11];
  const float* be2 = (const float*)d_in[12];
  const float* g3 = (const float*)d_in[13];
  const float* be3 = (const float*)d_in[14];
  const float* Wo = (const float*)d_in[15];
  const float* bo = (const float*)d_in[16];

  const int N = N_NODES_C;
  const int E = in_sizes[1] / 2;
  const int* src = ei;
  const int* dst = ei + E;

  // workspace carve-up
  float* AB   = (float*)d_ws;                         // [N,128]
  unsigned int* M = (unsigned int*)(AB + (size_t)N * 128);  // [N,64] keys; reused as feat buf
  float* Wcat = (float*)(M + (size_t)N * 64);         // [64,128]
  float* ssum = Wcat + 64 * 128;                      // [64]
  float* ssq  = ssum + 64;                            // [64]
  float* scale = ssq + 64;                            // [64]
  float* shift = scale + 64;                          // [64]
  float* gsum = shift + 64;                           // [64,64]
  float* gcnt = gsum + 64 * 64;                       // [64]

  const int gemmBlocks = N / 16;             // 6250
  const int edgeBlocks = (E + 15) / 16;      // 100000 (16 edges/block, 16 lanes/edge)
  const int elemBlocks = (N * 64) / 256;     // 25000

  const float* feat = x;
  const float* Ws[3]   = {W1, W2, W3};
  const float* bs[3]   = {b1, b2, b3};
  const float* gms[3]  = {g1, g2, g3};
  const float* bts[3]  = {be1, be2, be3};

  for (int layer = 0; layer < 3; ++layer) {
    prep_w_kernel<<<32, 256, 0, stream>>>(Ws[layer], Wcat);
    node_gemm_kernel<<<gemmBlocks, 256, 0, stream>>>(feat, Wcat, AB);
    zero_kernel<<<4096, 256, 0, stream>>>(M, N * 64);
    zero_kernel<<<1, 128, 0, stream>>>((unsigned int*)ssum, 128);  // ssum + ssq contiguous
    edge_max_kernel<<<edgeBlocks, 256, 0, stream>>>(src, dst, AB, M, E);
    combine_stats_kernel<<<512, 256, 0, stream>>>(AB, M, bs[layer], ssum, ssq, N);
    bn_finalize_kernel<<<1, 64, 0, stream>>>(ssum, ssq, gms[layer], bts[layer], scale, shift, N);
    if (layer < 2) {
      bn_relu_kernel<<<elemBlocks, 256, 0, stream>>>(AB, scale, shift, (float*)M, N);
      feat = (const float*)M;
    }
  }

  // pooling + head (layer-3 BN affine folded into pooled mean)
  zero_kernel<<<17, 256, 0, stream>>>((unsigned int*)gsum, 64 * 64 + 64);  // gsum+gcnt contiguous
  pool_kernel<<<1024, 256, 0, stream>>>(AB, batch, gsum, gcnt, N);
  head_kernel<<<1, 64, 0, stream>>>(gsum, gcnt, scale, shift, Wo, bo, (float*)d_out);
}